// AttentionLayer_37580963840428
// MI455X (gfx1250) — compile-verified
//
#include <hip/hip_runtime.h>
#include <hip/hip_bf16.h>
#include <stddef.h>

// Problem constants (match the reference exactly).
#define BB 1024
#define NN 100000
#define DD 8

#define NTILES 64      // BB/16 : 16 batch rows per WMMA tile
#define NSPLIT 25      // N-dimension splits (cross-workgroup)
#define CHUNK  4000    // NN / NSPLIT : j-range per workgroup
#define WCHUNK 500     // CHUNK / 8 waves : j-range per wave
#define KUNROLL 5      // WMMAs per unrolled trip (20 j-values; 500 = 25*20)

typedef __attribute__((ext_vector_type(2))) float v2f;
typedef __attribute__((ext_vector_type(8))) float v8f;

// ---------------------------------------------------------------------------
// Kernel 1: per-item score  s[j] = relu(gamma[j,:] @ Wa + ba0) @ h + ba1
// ---------------------------------------------------------------------------
__global__ void score_kernel(const float* __restrict__ gamma,
                             const float* __restrict__ Wa,
                             const float* __restrict__ ba0,
                             const float* __restrict__ ba1,
                             const float* __restrict__ h,
                             float* __restrict__ s) {
    int j = blockIdx.x * blockDim.x + threadIdx.x;
    if (j >= NN) return;
    float g[DD];
#pragma unroll
    for (int i = 0; i < DD; ++i) g[i] = gamma[(size_t)j * DD + i];
    float acc = ba1[0];
#pragma unroll
    for (int k = 0; k < DD; ++k) {
        float c = ba0[k];
#pragma unroll
        for (int i = 0; i < DD; ++i) c = fmaf(g[i], Wa[i * DD + k], c);
        c = fmaxf(c, 0.0f);
        acc = fmaf(c, h[k], acc);
    }
    s[j] = acc;
}

// ---------------------------------------------------------------------------
// Kernel 2: fused  E = exp(inputs * s);  partial[tile,split] = E_chunk @ [gamma | 1]
// V_WMMA_F32_16X16X4_F32 per 4 j-values; B-matrix column 8 is 1.0 so the
// accumulator column 8 carries sum(E) (softmax denominator) for free.
// All loads are unconditional (invalid B lanes clamped to column 0, value
// replaced by cndmask) -> no exec divergence in the hot loop.
// ---------------------------------------------------------------------------
__global__ __launch_bounds__(256) void attn_kernel(const float* __restrict__ inputs,
                                                   const float* __restrict__ gamma,
                                                   const float* __restrict__ s,
                                                   float* __restrict__ parts) {
    __shared__ float red[8 * 256];

    const int tile  = blockIdx.x;           // 0..63  : 16 batch rows each
    const int split = blockIdx.y;           // 0..24  : j-range chunk
    const int wave  = threadIdx.x >> 5;     // 0..7
    const int lane  = threadIdx.x & 31;

    // A-operand layout (32-bit A 16x4): lanes 0-15 hold K=0,1; lanes 16-31 K=2,3.
    const int r    = lane & 15;             // matrix row (M) / B column (N)
    const int koff = (lane >> 4) * 2;       // K offset within the 4-wide step

    const int  jbeg   = split * CHUNK + wave * WCHUNK;
    const bool gvalid = (r < DD);                   // B columns 0..7 from gamma
    const float bpad  = (r == DD) ? 1.0f : 0.0f;    // column 8 = ones -> sumE

    // Hoisted base pointers; inner loop uses immediate offsets only.
    const float* pIn = inputs + (size_t)(tile * 16 + r) * NN + jbeg + koff;
    const float* pS  = s + jbeg + koff;
    const float* pG  = gamma + (size_t)(jbeg + koff) * DD + (gvalid ? r : 0);

    v8f acc = {0.f, 0.f, 0.f, 0.f, 0.f, 0.f, 0.f, 0.f};

    for (int it = 0; it < WCHUNK / (4 * KUNROLL); ++it) {
        __builtin_prefetch(pIn + 4 * KUNROLL * 8, 0, 0);   // stream inputs ahead
#pragma unroll
        for (int u = 0; u < KUNROLL; ++u) {
            v2f in = *(const v2f*)(pIn + u * 4);           // inputs[m, j+koff..+1]
            v2f sv = *(const v2f*)(pS + u * 4);            // s[j+koff..+1]
            float g0 = pG[u * 4 * DD];                     // gamma[j+koff,   r]
            float g1 = pG[u * 4 * DD + DD];                // gamma[j+koff+1, r]
            v2f a;
            a.x = __expf(in.x * sv.x);
            a.y = __expf(in.y * sv.y);
            v2f b;
            b.x = gvalid ? g0 : bpad;
            b.y = gvalid ? g1 : bpad;
            acc = __builtin_amdgcn_wmma_f32_16x16x4_f32(
                      /*neg_a=*/false, a, /*neg_b=*/false, b,
                      /*c_mod=*/(short)0, acc, /*reuse_a=*/false, /*reuse_b=*/false);
        }
        pIn += 4 * KUNROLL;
        pS  += 4 * KUNROLL;
        pG  += 4 * KUNROLL * DD;
    }

    // Cross-wave reduction of the 16x16 f32 accumulator tile via LDS.
#pragma unroll
    for (int i = 0; i < 8; ++i) red[wave * 256 + i * 32 + lane] = acc[i];
    __syncthreads();

    const int tid = threadIdx.x;
    float sum = 0.f;
#pragma unroll
    for (int w = 0; w < 8; ++w) sum += red[w * 256 + tid];

    // C/D layout: VGPR i, lane l  ->  M = i + (l&16 ? 8 : 0), N = l&15
    const int i = tid >> 5;
    const int l = tid & 31;
    const int m = i + ((l & 16) ? 8 : 0);
    const int n = l & 15;
    if (n <= DD) {   // columns 0..7 = out_raw, column 8 = sumE
        parts[((size_t)(tile * NSPLIT + split)) * 144 + m * 9 + n] = sum;
    }
}

// ---------------------------------------------------------------------------
// Kernel 3: reduce the 25 N-splits and apply the softmax denominator.
// ---------------------------------------------------------------------------
__global__ void finalize_kernel(const float* __restrict__ parts,
                                float* __restrict__ out) {
    int t = blockIdx.x * blockDim.x + threadIdx.x;   // 0 .. 8191
    if (t >= BB * DD) return;
    const int b = t >> 3, d = t & 7;
    const int tile = b >> 4, m = b & 15;
    float num = 0.f, den = 0.f;
    for (int sp = 0; sp < NSPLIT; ++sp) {
        const float* p = parts + ((size_t)(tile * NSPLIT + sp)) * 144 + m * 9;
        num += p[d];
        den += p[8];
    }
    out[t] = num / den;
}

// ---------------------------------------------------------------------------
extern "C" void kernel_launch(void* const* d_in, const int* in_sizes, int n_in,
                              void* d_out, int out_size, void* d_ws, size_t ws_size,
                              hipStream_t stream) {
    const float* inputs = (const float*)d_in[0];   // [B, N]
    const float* gamma  = (const float*)d_in[1];   // [N, 8]
    const float* Wa     = (const float*)d_in[2];   // [8, 8]
    const float* ba0    = (const float*)d_in[3];   // [8]
    const float* ba1    = (const float*)d_in[4];   // [1]
    const float* h      = (const float*)d_in[5];   // [8, 1]
    float* out = (float*)d_out;                    // [B, 8]

    // Workspace layout (floats): s[NN] at 0; partials at 100096 (128B aligned).
    float* s     = (float*)d_ws;
    float* parts = (float*)d_ws + 100096;          // 64*25*144 = 230400 floats

    score_kernel<<<(NN + 255) / 256, 256, 0, stream>>>(gamma, Wa, ba0, ba1, h, s);

    dim3 grid(NTILES, NSPLIT);
    attn_kernel<<<grid, 256, 0, stream>>>(inputs, gamma, s, parts);

    finalize_kernel<<<(BB * DD + 255) / 256, 256, 0, stream>>>(parts, out);
}